// GIN_64209761075688
// MI455X (gfx1250) — compile-verified
//
#include <hip/hip_runtime.h>

// ---------------------------------------------------------------------------
// GIN forward on gfx1250: scatter-sum aggregation + f32 WMMA (16x16x4) MLPs.
// Bandwidth-bound problem (~0.8 GB traffic); WMMA makes GEMM compute free and
// keeps full fp32 precision (accumulate in f32, no bf16 rounding risk).
// ---------------------------------------------------------------------------

typedef float v2f __attribute__((ext_vector_type(2)));
typedef float v8f __attribute__((ext_vector_type(8)));

#define HDIM 64

__device__ __forceinline__ v8f wmma_f32_4(v2f a, v2f b, v8f c) {
  // D = A(16x4,f32) * B(4x16,f32) + C(16x16,f32)
  return __builtin_amdgcn_wmma_f32_16x16x4_f32(
      /*neg_a=*/false, a, /*neg_b=*/false, b,
      /*c_mod=*/(short)0, c, /*reuse_a=*/false, /*reuse_b=*/false);
}

// ----------------------------- utility kernels -----------------------------

__global__ void copy4_kernel(float4* __restrict__ dst, const float4* __restrict__ src, int n4) {
  int i = blockIdx.x * blockDim.x + threadIdx.x;
  if (i < n4) dst[i] = src[i];
}

__global__ void zero_kernel(float* __restrict__ p, int n) {
  int i = blockIdx.x * blockDim.x + threadIdx.x;
  if (i < n) p[i] = 0.0f;
}

// z[tgt,:] += h[src,:] for every edge; one thread per (edge, 4-feature chunk)
__global__ void scatter_add_kernel(float* __restrict__ z, const float* __restrict__ h,
                                   const long long* __restrict__ ei, int E, int D) {
  const int cpe = D >> 2;  // float4 chunks per edge
  int tid = blockIdx.x * blockDim.x + threadIdx.x;
  if (tid >= E * cpe) return;
  int e = tid / cpe;
  int c = tid - e * cpe;
  const long long src = ei[e];             // row 0 of edge_index
  const long long tgt = ei[(size_t)E + e]; // row 1 of edge_index
  const float4 v = *(const float4*)(h + (size_t)src * D + c * 4);
  float* p = z + (size_t)tgt * D + c * 4;
  atomicAdd(p + 0, v.x);
  atomicAdd(p + 1, v.y);
  atomicAdd(p + 2, v.z);
  atomicAdd(p + 3, v.w);
}

// pooled[batch[n],:] += h2[n,:]; counts[batch[n]] += 1
__global__ void pool_scatter_kernel(const float* __restrict__ h2, const long long* __restrict__ batch,
                                    float* __restrict__ pooled, float* __restrict__ counts, int N) {
  int tid = blockIdx.x * blockDim.x + threadIdx.x;
  if (tid >= N * 16) return;
  int n = tid >> 4;
  int c = tid & 15;
  const long long g = batch[n];
  const float4 v = *(const float4*)(h2 + (size_t)n * HDIM + c * 4);
  float* p = pooled + (size_t)g * HDIM + c * 4;
  atomicAdd(p + 0, v.x);
  atomicAdd(p + 1, v.y);
  atomicAdd(p + 2, v.z);
  atomicAdd(p + 3, v.w);
  if (c == 0) atomicAdd(counts + g, 1.0f);
}

// out[g] = dot(pooled[g]/max(cnt,1), fcW) + fcb  -- one wave32 per graph
__global__ void pool_final_kernel(const float* __restrict__ pooled, const float* __restrict__ counts,
                                  const float* __restrict__ fcW, const float* __restrict__ fcb,
                                  float* __restrict__ out, int G) {
  int gw = (blockIdx.x * blockDim.x + threadIdx.x) >> 5;
  int lane = threadIdx.x & 31;
  if (gw >= G) return;
  float s = pooled[(size_t)gw * HDIM + lane]      * fcW[lane] +
            pooled[(size_t)gw * HDIM + 32 + lane] * fcW[32 + lane];
#pragma unroll
  for (int off = 16; off > 0; off >>= 1) s += __shfl_down(s, off, 32);
  if (lane == 0) {
    float c = counts[gw];
    out[gw] = s / fmaxf(c, 1.0f) + fcb[0];
  }
}

// -------------------- fused 2-layer MLP via f32 WMMA -----------------------
// Hout = relu( relu(Z @ Wa + ba) @ Wb + bb ), Z:[N,DIN], Wa:[DIN,64], Wb:[64,64]
// One wave per 16-node tile; 4 column tiles of 16 cover H=64.
template <int DIN>
__global__ __launch_bounds__(128) void gin_mlp_kernel(
    const float* __restrict__ Z,
    const float* __restrict__ Wa, const float* __restrict__ ba,
    const float* __restrict__ Wb, const float* __restrict__ bb,
    float* __restrict__ Hout, int N) {
  __shared__ float hid[4][16][HDIM + 4];  // per-wave staging, padded stride

  const int wave = threadIdx.x >> 5;
  const int lane = threadIdx.x & 31;
  const int sub  = lane >> 4;   // half-wave: selects K pair (A) / K rows (B)
  const int mrow = lane & 15;
  const int tile = blockIdx.x * 4 + wave;
  const int m0   = tile * 16;
  const bool valid = (m0 < N);  // wave-uniform -> EXEC stays all-ones

  // ---- GEMM 1: hidden = relu(Z_tile @ Wa + ba) ----
  v8f acc[4] = {};
  if (valid) {
    const float* zrow = Z + (size_t)(m0 + mrow) * DIN;
#pragma unroll 4
    for (int k = 0; k < DIN; k += 4) {
      const float4 av = *(const float4*)(zrow + k);
      v2f A;
      A.x = sub ? av.z : av.x;  // K = k + 2*sub
      A.y = sub ? av.w : av.y;  // K = k + 1 + 2*sub
      const int kb = k + 2 * sub;
#pragma unroll
      for (int t = 0; t < 4; ++t) {
        const int col = t * 16 + mrow;
        v2f B;
        B.x = Wa[(size_t)kb * HDIM + col];
        B.y = Wa[(size_t)(kb + 1) * HDIM + col];
        acc[t] = wmma_f32_4(A, B, acc[t]);
      }
    }
    float bav[4];
#pragma unroll
    for (int t = 0; t < 4; ++t) bav[t] = ba[t * 16 + mrow];
#pragma unroll
    for (int r = 0; r < 8; ++r) {
      const int row = r + 8 * sub;  // C layout: lanes 0-15 -> M=r, 16-31 -> M=8+r
#pragma unroll
      for (int t = 0; t < 4; ++t)
        hid[wave][row][t * 16 + mrow] = fmaxf(acc[t][r] + bav[t], 0.0f);
    }
  }
  __syncthreads();  // uniform trip count across all waves of the block

  // ---- GEMM 2: out = relu(hidden @ Wb + bb) ----
  if (valid) {
    v8f o[4] = {};
#pragma unroll 4
    for (int k = 0; k < HDIM; k += 4) {
      const int kb = k + 2 * sub;
      v2f A;
      A.x = hid[wave][mrow][kb];
      A.y = hid[wave][mrow][kb + 1];
#pragma unroll
      for (int t = 0; t < 4; ++t) {
        const int col = t * 16 + mrow;
        v2f B;
        B.x = Wb[(size_t)kb * HDIM + col];
        B.y = Wb[(size_t)(kb + 1) * HDIM + col];
        o[t] = wmma_f32_4(A, B, o[t]);
      }
    }
    float bbv[4];
#pragma unroll
    for (int t = 0; t < 4; ++t) bbv[t] = bb[t * 16 + mrow];
#pragma unroll
    for (int r = 0; r < 8; ++r) {
      float* orow = Hout + (size_t)(m0 + r + 8 * sub) * HDIM;
#pragma unroll
      for (int t = 0; t < 4; ++t)
        orow[t * 16 + mrow] = fmaxf(o[t][r] + bbv[t], 0.0f);
    }
  }
}

// ------------------------------- launcher ----------------------------------

extern "C" void kernel_launch(void* const* d_in, const int* in_sizes, int n_in,
                              void* d_out, int out_size, void* d_ws, size_t ws_size,
                              hipStream_t stream) {
  const float*     x     = (const float*)d_in[0];
  const long long* ei    = (const long long*)d_in[1];   // [2, E] int64
  const long long* batch = (const long long*)d_in[2];   // [N] int64
  const float* W1a = (const float*)d_in[3];
  const float* b1a = (const float*)d_in[4];
  const float* W1b = (const float*)d_in[5];
  const float* b1b = (const float*)d_in[6];
  const float* W2a = (const float*)d_in[7];
  const float* b2a = (const float*)d_in[8];
  const float* W2b = (const float*)d_in[9];
  const float* b2b = (const float*)d_in[10];
  const float* fcW = (const float*)d_in[11];
  const float* fcb = (const float*)d_in[12];
  float* out = (float*)d_out;

  const int N = in_sizes[0] / 128;  // F_IN = 128
  const int E = in_sizes[1] / 2;
  const int G = out_size;

  // workspace layout (floats): z1[N*128] | h1[N*64] | z2[N*64] | pooled[G*64] | counts[G]
  float* ws = (float*)d_ws;
  float* z1 = ws;
  float* h1 = ws + (size_t)N * 128;
  float* z2 = ws + (size_t)N * 192;
  float* h2 = z1;  // z1 dead after MLP1 consumes it -> reuse for h2
  float* pooled = ws + (size_t)N * 256;
  float* counts = pooled + (size_t)G * HDIM;

  const int TB = 256;

  // Layer 1: z1 = x + scatter_sum(x)
  { int n4 = N * 32;
    copy4_kernel<<<(n4 + TB - 1) / TB, TB, 0, stream>>>((float4*)z1, (const float4*)x, n4); }
  { int nt = E * 32;
    scatter_add_kernel<<<(nt + TB - 1) / TB, TB, 0, stream>>>(z1, x, ei, E, 128); }
  gin_mlp_kernel<128><<<(N + 63) / 64, 128, 0, stream>>>(z1, W1a, b1a, W1b, b1b, h1, N);

  // Layer 2: z2 = h1 + scatter_sum(h1)
  { int n4 = N * 16;
    copy4_kernel<<<(n4 + TB - 1) / TB, TB, 0, stream>>>((float4*)z2, (const float4*)h1, n4); }
  { int nt = E * 16;
    scatter_add_kernel<<<(nt + TB - 1) / TB, TB, 0, stream>>>(z2, h1, ei, E, 64); }
  gin_mlp_kernel<64><<<(N + 63) / 64, 128, 0, stream>>>(z2, W2a, b2a, W2b, b2b, h2, N);

  // Mean pool + FC head
  { int n = G * (HDIM + 1);
    zero_kernel<<<(n + TB - 1) / TB, TB, 0, stream>>>(pooled, n); }
  { int nt = N * 16;
    pool_scatter_kernel<<<(nt + TB - 1) / TB, TB, 0, stream>>>(h2, batch, pooled, counts, N); }
  { int nt = G * 32;
    pool_final_kernel<<<(nt + TB - 1) / TB, TB, 0, stream>>>(pooled, counts, fcW, fcb, out, G); }

  (void)n_in; (void)ws_size; (void)b1a; (void)b2a;
}